// Cont_Loss_51213190037601
// MI455X (gfx1250) — compile-verified
//
#include <hip/hip_runtime.h>

#define S 64
#define NROWS (2*8*32*32*32)   // 524288 rows of X[N, 64]
#define NSTEPS (NROWS/4)       // K-steps of 4 rows each
#define MARGIN 60000.0f

typedef __attribute__((ext_vector_type(2))) float v2f;
typedef __attribute__((ext_vector_type(8))) float v8f;

// ---------------------------------------------------------------------------
// Kernel A: partial Gram matrices via V_WMMA_F32_16X16X4_F32.
// Each wave owns 10 (ta>=tb) 16x16 fp32 accumulator tiles and grid-strides
// over K-chunks of 4 rows. Cross-wave combine via LDS ds_add_f32, then the
// workgroup writes its 64x64 partial to d_ws (deterministic, no global atomics).
// ---------------------------------------------------------------------------
__global__ void __launch_bounds__(256)
gram_partial_kernel(const float* __restrict__ x, float* __restrict__ partial, int nwg) {
    __shared__ float Gs[S * S];
    const int tid = threadIdx.x;
    for (int i = tid; i < S * S; i += 256) Gs[i] = 0.0f;
    __syncthreads();

    const int lane = tid & 31;
    const int wave = tid >> 5;        // 0..7
    const int half = lane >> 4;       // 0 | 1 (selects K=0,1 vs K=2,3 rows)
    const int l    = lane & 15;       // time index within tile

    v8f zero = {};
    v8f acc[10];
#pragma unroll
    for (int i = 0; i < 10; ++i) acc[i] = zero;

    const int totalWaves = nwg * 8;
    const int gwave = blockIdx.x * 8 + wave;

    for (int step = gwave; step < NSTEPS; step += totalWaves) {
        const size_t n0 = (size_t)step * 4;
        // Fragment for tile j holds X[n0+2*half + {0,1}, 16*j + l]
        const float* rowp = x + (n0 + (size_t)(2 * half)) * S + l;
        v2f f[4];
#pragma unroll
        for (int j = 0; j < 4; ++j) {
            f[j].x = rowp[16 * j];        // VGPR0: K = 2*half
            f[j].y = rowp[16 * j + S];    // VGPR1: K = 2*half + 1
        }
        // Same fragment serves as A (row tile) and B (col tile).
        int t = 0;
#pragma unroll
        for (int ta = 0; ta < 4; ++ta) {
#pragma unroll
            for (int tb = 0; tb <= ta; ++tb, ++t) {
                acc[t] = __builtin_amdgcn_wmma_f32_16x16x4_f32(
                    false, f[ta], false, f[tb], (short)0, acc[t], false, false);
            }
        }
    }

    // LDS combine across the 8 waves (ds_add_f32).
    {
        int t = 0;
#pragma unroll
        for (int ta = 0; ta < 4; ++ta) {
#pragma unroll
            for (int tb = 0; tb <= ta; ++tb, ++t) {
#pragma unroll
                for (int g = 0; g < 8; ++g) {
                    const int a = 16 * ta + g + 8 * half;   // C layout: VGPR g -> M = g (+8 for hi half)
                    const int b = 16 * tb + l;              // N = lane%16
                    atomicAdd(&Gs[a * S + b], acc[t][g]);
                }
            }
        }
    }
    __syncthreads();

    float* out = partial + (size_t)blockIdx.x * (S * S);
    for (int i = tid; i < S * S; i += 256) out[i] = Gs[i];
}

// ---------------------------------------------------------------------------
// Kernel B: column-sum the per-workgroup partials into the final Gram matrix.
// ---------------------------------------------------------------------------
__global__ void __launch_bounds__(256)
gram_reduce_kernel(const float* __restrict__ partial, float* __restrict__ G, int nwg) {
    const int g = blockIdx.x * 256 + threadIdx.x;   // 0..4095
    float s = 0.0f;
    for (int p = 0; p < nwg; ++p) s += partial[(size_t)p * (S * S) + g];
    G[g] = s;
}

// ---------------------------------------------------------------------------
// Kernel C: contrastive hinge loss from the Gram matrix.
// sq[t] = G[t][t];  D[a][b] = G[a][a] + G[b][b] - 2*G[a][b]
// ---------------------------------------------------------------------------
__global__ void __launch_bounds__(256)
loss_kernel(const float* __restrict__ G, float* __restrict__ out) {
    __shared__ float red[256];
    const int tid = threadIdx.x;
    float s = 0.0f;
    for (int i = tid; i < S * S; i += 256) {
        const int a = i >> 6;
        const int b = i & 63;
        if (a > b) {   // lower triangle only (all such entries live in computed tiles)
            const float d = G[a * S + a] + G[b * S + b] - 2.0f * G[a * S + b];
            s += (a - b == 1) ? fmaxf(0.0f, MARGIN - d) : d;
        }
    }
    red[tid] = s;
    __syncthreads();
    for (int w = 128; w > 0; w >>= 1) {
        if (tid < w) red[tid] += red[tid + w];
        __syncthreads();
    }
    if (tid == 0) out[0] = red[0] / (float)(S * (S - 1) * 1000);
}

// ---------------------------------------------------------------------------
extern "C" void kernel_launch(void* const* d_in, const int* in_sizes, int n_in,
                              void* d_out, int out_size, void* d_ws, size_t ws_size,
                              hipStream_t stream) {
    (void)in_sizes; (void)n_in; (void)out_size;
    const float* x = (const float*)d_in[0];
    float* out = (float*)d_out;

    // Scratch layout: [nwg * 4096 floats partials][4096 floats reduced G]
    const size_t tileBytes = (size_t)(S * S) * sizeof(float);
    int nwg = 1;
    if (ws_size > tileBytes) {
        size_t maxwg = (ws_size - tileBytes) / tileBytes;
        nwg = (int)(maxwg < 1 ? 1 : (maxwg > 256 ? 256 : maxwg));
    }
    float* partial = (float*)d_ws;
    float* G = partial + (size_t)nwg * (S * S);

    gram_partial_kernel<<<nwg, 256, 0, stream>>>(x, partial, nwg);
    gram_reduce_kernel<<<16, 256, 0, stream>>>(partial, G, nwg);
    loss_kernel<<<1, 256, 0, stream>>>(G, out);
}